// RelationalDynamicAggregation_33406255628295
// MI455X (gfx1250) — compile-verified
//
#include <hip/hip_runtime.h>
#include <math.h>

typedef __attribute__((ext_vector_type(2))) float v2f;
typedef __attribute__((ext_vector_type(8))) float v8f;

#define NB 256   // batch
#define NT 8     // targets (== waves per block)
#define NS 200   // sequence
#define SP 208   // sequence padded to multiple of 16
#define NR 8     // relations
#define ND 64    // feature dim
#define NF 20    // freqs

__global__ __launch_bounds__(256) void rda_fused_kernel(
    const float* __restrict__ seq,          // [B,S,D]
    const float* __restrict__ delta_t_n,    // [B,S]
    const float* __restrict__ target,       // [B,T,D]
    const float* __restrict__ target_value, // [B,T,R,D]
    const int*   __restrict__ valid_mask,   // [B,1,S,1] -> [B,S]
    const float* __restrict__ rel_emb,      // [R,D]
    const float* __restrict__ freq_real,    // [R,F]
    const float* __restrict__ freq_imag,    // [R,F]
    float* __restrict__ out)                // [B,T,R,D]
{
    __shared__ float sSeq[SP * ND];        // 53,248 B (zero-padded rows 200..207)
    __shared__ float sAtt[NT][SP * NR];    // 53,248 B (per-wave att / weights)
    __shared__ float sRi [NT][NR * ND];    // 16,384 B
    __shared__ float sDecay[SP * NR];      //  6,656 B
    __shared__ int   sValid[SP];           //    832 B

    const int b    = blockIdx.x;
    const int tid  = threadIdx.x;
    const int lane = tid & 31;
    const int w    = tid >> 5;             // wave id == target index t
    const int l15  = lane & 15;
    const int half = lane >> 4;            // 0: K+0/K+1 lanes, 1: K+2/K+3 lanes
    const int koff = half * 2;
    const bool loR = (l15 < NR);           // lane maps to a real r (vs zero pad)
    const int  rClamp = l15 & 7;           // clamped r index for unconditional loads

    // ---------------- Phase 0: stage seq, valid, decay into LDS ----------------
    const float* seqB = seq + (size_t)b * NS * ND;
    for (int idx = tid; idx < SP * (ND / 4); idx += 256) {
        int s = idx / (ND / 4);
        int c = idx % (ND / 4);
        float4 v = make_float4(0.f, 0.f, 0.f, 0.f);
        if (s < NS) v = ((const float4*)(seqB + s * ND))[c];
        ((float4*)sSeq)[idx] = v;
    }
    for (int s = tid; s < SP; s += 256)
        sValid[s] = (s < NS) ? valid_mask[b * NS + s] : 0;

    const float PI = 3.14159265358979323846f;
    for (int idx = tid; idx < SP * NR; idx += 256) {
        int s = idx / NR;
        int r = idx % NR;
        float dv = 0.f;
        if (s < NS && valid_mask[b * NS + s] != 0) {
            float dt  = delta_t_n[b * NS + s];
            float acc = 0.f;
            #pragma unroll
            for (int k = 0; k < NF; ++k) {
                float wk = (PI * (float)k / (float)(NF - 1)) * dt;
                float sw, cw;
                sincosf(wk, &sw, &cw);
                acc += cw * freq_real[r * NF + k] - sw * freq_imag[r * NF + k];
            }
            dv = acc * (1.0f / (2.0f * (float)NF));
            dv = fminf(fmaxf(dv, 0.0f), 1.0f);
        }
        sDecay[idx] = dv;
    }

    // ---------------- Phase 1: per-wave ri = (rel_emb + tv) * target ----------------
    const int t = w;
    {
        const float* tgt = target + ((size_t)b * NT + t) * ND;
        const float* tv  = target_value + ((size_t)b * NT + t) * NR * ND;
        for (int i = lane; i < NR * ND; i += 32) {
            int d = i & (ND - 1);
            sRi[t][i] = (rel_emb[i] + tv[i]) * tgt[d];
        }
    }
    __syncthreads();

    // ---------------- Phase 2: att[s,r] = seq[S×64] × ri^T[64×R]  (WMMA f32 16x16x4) ----------------
    float* attW = sAtt[w];
    const float* riW = sRi[w];

    // Hoist B fragments: ri is invariant across all 13 S-tiles.
    // Lanes with l15>=8 map to zero-pad columns -> unconditional clamped load + select.
    v2f riB[ND / 4];
    #pragma unroll
    for (int kk = 0; kk < ND / 4; ++kk) {
        int d0 = kk * 4 + koff;
        float b0 = riW[rClamp * ND + d0];
        float b1 = riW[rClamp * ND + d0 + 1];
        v2f Bv;
        Bv.x = loR ? b0 : 0.f;
        Bv.y = loR ? b1 : 0.f;
        riB[kk] = Bv;
    }

    for (int st = 0; st < SP / 16; ++st) {
        v8f acc = {};
        const int    row  = st * 16 + l15;         // M = s
        const float* aRow = sSeq + row * ND;
        #pragma unroll
        for (int kk = 0; kk < ND / 4; ++kk) {
            int d0 = kk * 4 + koff;
            v2f A = { aRow[d0], aRow[d0 + 1] };    // A: M=s rows, K=d
            acc = __builtin_amdgcn_wmma_f32_16x16x4_f32(
                false, A, false, riB[kk], (short)0, acc, false, false);
        }
        if (loR) {                                  // D: vgpr vi -> M=vi (+8 upper half), N=l15=r
            int sbase = st * 16 + half * 8;
            #pragma unroll
            for (int vi = 0; vi < 8; ++vi)
                attW[(sbase + vi) * NR + l15] = acc[vi];
        }
    }
    __syncthreads();

    // ---------------- Phase 3: masked softmax over S, fold in decay ----------------
    {
        const int r      = lane & 7;               // 4 lanes cooperate per column r
        const int rowOff = lane >> 3;              // 0..3
        float mx = -INFINITY;
        for (int s = rowOff; s < NS; s += 4)
            if (sValid[s]) mx = fmaxf(mx, attW[s * NR + r]);
        mx = fmaxf(mx, __shfl_xor(mx, 8, 32));
        mx = fmaxf(mx, __shfl_xor(mx, 16, 32));

        float sum = 0.f;
        for (int s = rowOff; s < NS; s += 4)
            if (sValid[s]) sum += __expf(attW[s * NR + r] - mx);
        sum += __shfl_xor(sum, 8, 32);
        sum += __shfl_xor(sum, 16, 32);
        float inv = 1.0f / sum;

        for (int s = rowOff; s < SP; s += 4) {
            float v = 0.f;
            if (s < NS && sValid[s])
                v = __expf(attW[s * NR + r] - mx) * inv * sDecay[s * NR + r];
            attW[s * NR + r] = v;
        }
    }
    __syncthreads();

    // ---------------- Phase 4: context[r,d] = att^T[R×S] × seq[S×64] ----------------
    v8f c0 = {}, c1 = {}, c2 = {}, c3 = {};
    for (int ks = 0; ks < SP / 4; ++ks) {
        int s0 = ks * 4 + koff;                    // K = s
        // A: M=r rows (8 real + 8 zero pad) -> clamped unconditional loads + select
        float a0 = attW[s0 * NR + rClamp];
        float a1 = attW[(s0 + 1) * NR + rClamp];
        v2f A;
        A.x = loR ? a0 : 0.f;
        A.y = loR ? a1 : 0.f;
        const float* p0 = sSeq + s0 * ND;          // B: N=d cols, 4 chunks of 16
        const float* p1 = sSeq + (s0 + 1) * ND;
        v2f B0 = { p0[l15],      p1[l15]      };
        v2f B1 = { p0[16 + l15], p1[16 + l15] };
        v2f B2 = { p0[32 + l15], p1[32 + l15] };
        v2f B3 = { p0[48 + l15], p1[48 + l15] };
        c0 = __builtin_amdgcn_wmma_f32_16x16x4_f32(false, A, false, B0, (short)0, c0, false, false);
        c1 = __builtin_amdgcn_wmma_f32_16x16x4_f32(false, A, false, B1, (short)0, c1, false, false);
        c2 = __builtin_amdgcn_wmma_f32_16x16x4_f32(false, A, false, B2, (short)0, c2, false, false);
        c3 = __builtin_amdgcn_wmma_f32_16x16x4_f32(false, A, false, B3, (short)0, c3, false, false);
    }

    // D rows 0..7 = real r, live in lanes 0..15 (vgpr vi -> r=vi), N=lane -> d
    if (lane < 16) {
        float* outp = out + ((size_t)b * NT + t) * NR * ND;
        #pragma unroll
        for (int vi = 0; vi < 8; ++vi) {
            outp[vi * ND +  0 + lane] = c0[vi];
            outp[vi * ND + 16 + lane] = c1[vi];
            outp[vi * ND + 32 + lane] = c2[vi];
            outp[vi * ND + 48 + lane] = c3[vi];
        }
    }
}

extern "C" void kernel_launch(void* const* d_in, const int* in_sizes, int n_in,
                              void* d_out, int out_size, void* d_ws, size_t ws_size,
                              hipStream_t stream) {
    const float* seq          = (const float*)d_in[0];
    const float* delta_t_n    = (const float*)d_in[1];
    const float* target       = (const float*)d_in[2];
    const float* target_value = (const float*)d_in[3];
    const int*   valid_mask   = (const int*)  d_in[4];
    const float* rel_emb      = (const float*)d_in[5];
    const float* freq_real    = (const float*)d_in[6];
    const float* freq_imag    = (const float*)d_in[7];
    float* out = (float*)d_out;

    rda_fused_kernel<<<NB, 256, 0, stream>>>(
        seq, delta_t_n, target, target_value, valid_mask,
        rel_emb, freq_real, freq_imag, out);
}